// ProjectLayer_10892037062829
// MI455X (gfx1250) — compile-verified
//
#include <hip/hip_runtime.h>

typedef __attribute__((ext_vector_type(2))) float v2f;
typedef __attribute__((ext_vector_type(8))) float v8f;

namespace {
constexpr int kN = 5, kB = 8, kC = 15, kH = 256, kW = 480;
constexpr int kCX = 80, kCY = 80, kCZ = 20;
constexpr int kBins = kCX * kCY * kCZ;   // 128000
constexpr int kHW = kH * kW;             // 122880
}

// Cross-half identity permute: lane L receives src from lane L^16.
// Pure VALU (v_permlanex16_b32) -- no LDS pipe, no DScnt waits.
__device__ __forceinline__ float permx16(float v) {
  return __int_as_float(__builtin_amdgcn_permlanex16(
      0, __float_as_int(v), 0x76543210u, 0xFEDCBA98u, false, false));
}

__global__ __launch_bounds__(256) void project_layer_kernel(
    const float* __restrict__ hmaps,   // [N,B,C,H,W]
    const float* __restrict__ Rm,      // [N,B,3,3]
    const float* __restrict__ Tv,      // [N,B,3]
    const float* __restrict__ fv,      // [N,B,2]
    const float* __restrict__ cv,      // [N,B,2]
    const float* __restrict__ trv,     // [N,B,2,3]
    const float* __restrict__ whv,     // [N,B,2]
    const float* __restrict__ gcv,     // [B,3]
    float* __restrict__ cubes,         // [B,C,NBINS]
    float* __restrict__ grids)         // [B,NBINS,3]
{
  const int b = blockIdx.y;
  const int p = blockIdx.x * blockDim.x + threadIdx.x;
  const int lane = threadIdx.x & 31;
  const bool lo = lane < 16;

  // voxel coordinates: p = (ix*80 + iy)*20 + iz
  const int iz  = p % kCZ;
  const int ixy = p / kCZ;
  const int iy  = ixy % kCY;
  const int ix  = ixy / kCY;

  const float gx = -4000.0f + (float)ix * (8000.0f / 79.0f) + gcv[b * 3 + 0];
  const float gy = -4000.0f + (float)iy * (8000.0f / 79.0f) + gcv[b * 3 + 1];
  const float gz = -1000.0f + (float)iz * (2000.0f / 19.0f) + gcv[b * 3 + 2];

  {
    // write-once output: non-temporal so heatmap gather set keeps L2
    float* g = grids + ((size_t)b * kBins + p) * 3;
    __builtin_nontemporal_store(gx, g + 0);
    __builtin_nontemporal_store(gy, g + 1);
    __builtin_nontemporal_store(gz, g + 2);
  }

  float acc[kC];
#pragma unroll
  for (int c = 0; c < kC; ++c) acc[c] = 0.0f;
  float cnt = 0.0f;

  for (int n = 0; n < kN; ++n) {
    const int nb = n * kB + b;                 // block-uniform -> scalar loads
    const float* Rp = Rm + nb * 9;
    const float* Tp = Tv + nb * 3;
    const float f0 = fv[nb*2+0], f1 = fv[nb*2+1];
    const float c0 = cv[nb*2+0], c1 = cv[nb*2+1];
    const float t00 = trv[nb*6+0], t01 = trv[nb*6+1], t02 = trv[nb*6+2];
    const float t10 = trv[nb*6+3], t11 = trv[nb*6+4], t12 = trv[nb*6+5];
    const float w0 = whv[nb*2+0], w1 = whv[nb*2+1];

    const float ptx = gx - Tp[0];
    const float pty = gy - Tp[1];
    const float ptz = gz - Tp[2];

    // ---- xcam = R * pt for all 32 lanes via two V_WMMA_F32_16X16X4_F32 ----
    // A (16x4): rows 0..2 = R, rest 0.
    // ISA layout: lane L<16 -> (A[L][0], A[L][1]); lane L>=16 -> (A[L-16][2], A[L-16][3]).
    const float r00=Rp[0], r01=Rp[1], r02=Rp[2];
    const float r10=Rp[3], r11=Rp[4], r12=Rp[5];
    const float r20=Rp[6], r21=Rp[7], r22=Rp[8];
    v2f A;
    A[0] = (lane==0)?r00:(lane==1)?r10:(lane==2)?r20:
           (lane==16)?r02:(lane==17)?r12:(lane==18)?r22:0.0f;
    A[1] = (lane==0)?r01:(lane==1)?r11:(lane==2)?r21:0.0f;

    // B (4x16): VGPR0 holds rows {0,2} (lanes 0-15 / 16-31), VGPR1 rows {1,3}.
    // Columns = points; WMMA#1 covers lanes 0-15's voxels, WMMA#2 lanes 16-31's.
    const float sx = permx16(ptx);   // lane L -> pt of lane L^16
    const float sy = permx16(pty);
    const float sz = permx16(ptz);
    v2f B1, B2;
    B1[0] = lo ? ptx : sz;   B1[1] = lo ? pty : 0.0f;
    B2[0] = lo ? sx  : ptz;  B2[1] = lo ? sy  : 0.0f;

    v8f czero = {0.0f,0.0f,0.0f,0.0f,0.0f,0.0f,0.0f,0.0f};
    v8f dA = __builtin_amdgcn_wmma_f32_16x16x4_f32(false, A, false, B1, (short)0, czero, false, false);
    v8f dB = __builtin_amdgcn_wmma_f32_16x16x4_f32(false, A, false, B2, (short)0, czero, false, false);

    // D rows 0..2 = xcam components; lanes 0-15 already hold their point's values,
    // lanes 16-31 fetch theirs from WMMA#2's lanes 0-15.
    // Evaluate the convergent permutes unconditionally (straight-line) so the
    // selects lower to v_cndmask instead of exec-masked regions.
    const float tx = permx16(dB[0]);
    const float ty = permx16(dB[1]);
    const float tz = permx16(dB[2]);
    const float xcx = lo ? dA[0] : tx;
    const float xcy = lo ? dA[1] : ty;
    const float xcz = lo ? dA[2] : tz;
    // -----------------------------------------------------------------------

    const float invz = 1.0f / xcz;
    float xq = f0 * (xcx * invz) + c0;
    float yq = f1 * (xcy * invz) + c1;

    const bool inb = (xq >= 0.0f) && (xq < w0) && (yq >= 0.0f) && (yq < w1);
    const float mx = fmaxf(w0, w1);
    xq = fminf(fmaxf(xq, -1.0f), mx);
    yq = fminf(fmaxf(yq, -1.0f), mx);

    const float xt = (t00 * xq + t01 * yq + t02) * 0.5f;   // * (HM/IMG) = 0.5
    const float yt = (t10 * xq + t11 * yq + t12) * 0.5f;

    const float sgx = fminf(fmaxf(xt * (2.0f / 479.0f) - 1.0f, -1.1f), 1.1f);
    const float sgy = fminf(fmaxf(yt * (2.0f / 255.0f) - 1.0f, -1.1f), 1.1f);

    const float fx = (sgx + 1.0f) * 0.5f * 479.0f;
    const float fy = (sgy + 1.0f) * 0.5f * 255.0f;
    const float x0 = floorf(fx), y0 = floorf(fy);
    const float x1 = x0 + 1.0f,  y1 = y0 + 1.0f;
    const float wx1 = fx - x0, wx0 = 1.0f - wx1;
    const float wy1 = fy - y0, wy0 = 1.0f - wy1;

    const float vx0 = (x0 >= 0.0f && x0 <= 479.0f) ? 1.0f : 0.0f;
    const float vx1 = (x1 >= 0.0f && x1 <= 479.0f) ? 1.0f : 0.0f;
    const float vy0 = (y0 >= 0.0f && y0 <= 255.0f) ? 1.0f : 0.0f;
    const float vy1 = (y1 >= 0.0f && y1 <= 255.0f) ? 1.0f : 0.0f;

    const int x0c = (int)fminf(fmaxf(x0, 0.0f), 479.0f);
    const int x1c = (int)fminf(fmaxf(x1, 0.0f), 479.0f);
    const int y0c = (int)fminf(fmaxf(y0, 0.0f), 255.0f);
    const int y1c = (int)fminf(fmaxf(y1, 0.0f), 255.0f);

    const float w00 = wx0 * wy0 * vx0 * vy0;
    const float w01 = wx1 * wy0 * vx1 * vy0;
    const float w10 = wx0 * wy1 * vx0 * vy1;
    const float w11 = wx1 * wy1 * vx1 * vy1;

    if (inb) {   // bnd==0 contributions are exactly zero -> predicate the gathers
      cnt += 1.0f;
      const float* base = hmaps + (size_t)nb * kC * kHW;
      const int o00 = y0c * kW + x0c, o01 = y0c * kW + x1c;
      const int o10 = y1c * kW + x0c, o11 = y1c * kW + x1c;
#pragma unroll
      for (int c = 0; c < kC; ++c) {
        const float* pc = base + (size_t)c * kHW;
        float v = w00 * pc[o00];
        v = fmaf(w01, pc[o01], v);
        v = fmaf(w10, pc[o10], v);
        v = fmaf(w11, pc[o11], v);
        acc[c] += v;
      }
    }
  }

  const float inv = 1.0f / (cnt + 1e-6f);
#pragma unroll
  for (int c = 0; c < kC; ++c) {
    float v = acc[c] * inv;
    v = fmaxf(v, 0.0f);   // NaN -> 0, matching nan_to_num + clip
    v = fminf(v, 1.0f);
    __builtin_nontemporal_store(v, cubes + ((size_t)b * kC + c) * kBins + p);
  }
}

extern "C" void kernel_launch(void* const* d_in, const int* in_sizes, int n_in,
                              void* d_out, int out_size, void* d_ws, size_t ws_size,
                              hipStream_t stream) {
  const float* hm = (const float*)d_in[0];
  const float* R  = (const float*)d_in[1];
  const float* T  = (const float*)d_in[2];
  const float* f  = (const float*)d_in[3];
  const float* c  = (const float*)d_in[4];
  const float* tr = (const float*)d_in[5];
  const float* wh = (const float*)d_in[6];
  const float* gc = (const float*)d_in[7];

  float* cubes = (float*)d_out;
  float* grids = cubes + (size_t)kB * kC * kBins;   // second tuple output

  dim3 grid(kBins / 256, kB, 1);
  project_layer_kernel<<<grid, dim3(256, 1, 1), 0, stream>>>(
      hm, R, T, f, c, tr, wh, gc, cubes, grids);
}